// evals_31198642438436
// MI455X (gfx1250) — compile-verified
//
#include <hip/hip_runtime.h>

// ---------------------------------------------------------------------------
// Confusion-matrix reduction: out[b,c] = [tp, tn, fp, fn]
//   tp = sum(p*t), sp = sum(p), st = sum(t)
//   fp = sp - tp, fn = st - tp, tn = N - sp - st + tp
// Memory-bound: 96 MB streamed once -> ~4.1 us at 23.3 TB/s.
// ---------------------------------------------------------------------------

typedef __attribute__((ext_vector_type(2))) float v2f;
typedef __attribute__((ext_vector_type(4))) float v4f;
typedef __attribute__((ext_vector_type(8))) float v8f;

#define PLANE_N (512 * 512)        // elements per (b,c) plane
#define BLOCKS_PER_PLANE 32
#define THREADS 256
#define CHUNK (PLANE_N / BLOCKS_PER_PLANE)          // 8192 floats per block
#define ITERS (CHUNK / (THREADS * 4))               // 8 float4 loads per thread

// Single-instruction lane <-> lane^16 exchange:
// DS_SWIZZLE_B32 group-of-32 mode, SWAPX16: xor=0x10, or=0x00, and=0x1f
//   offset = (xor<<10)|(or<<5)|and = 0x401F
__device__ __forceinline__ float swapx16(float s) {
    int si = __builtin_bit_cast(int, s);
    int so = __builtin_amdgcn_ds_swizzle(si, 0x401F);
    return __builtin_bit_cast(float, so);
}

// Full 32-lane sum using V_WMMA_F32_16X16X4_F32.
// A = (x, 0) per lane -> the 64 A-register slots hold each lane's partial once
// (plus zeros). B = all-ones, so D[m][n] = sum_k A[m][k] independent of B's
// internal layout. Summing the 8 C/D VGPRs per lane covers 8 of the 16 rows
// (lanes 0-15: rows 0-7, lanes 16-31: rows 8-15); adding the xor-16 partner
// covers all rows => total over all 64 A slots == sum of the 32 partials.
// Requires EXEC all-ones (callers launch full, divergence-free waves).
__device__ __forceinline__ float wave_sum32_wmma(float x) {
    v2f a;
    a[0] = x;
    a[1] = 0.0f;
    v2f b;
    b[0] = 1.0f;
    b[1] = 1.0f;
    v8f c = {};
    c = __builtin_amdgcn_wmma_f32_16x16x4_f32(
        /*neg_a=*/false, a, /*neg_b=*/false, b,
        /*c_mod=*/(short)0, c, /*reuse_a=*/false, /*reuse_b=*/false);
    float s = ((c[0] + c[1]) + (c[2] + c[3])) + ((c[4] + c[5]) + (c[6] + c[7]));
    return s + swapx16(s);  // full 32-lane sum, broadcast to every lane
}

// Kernel 1: streaming partial sums. One block covers CHUNK contiguous floats
// of one plane; writes (sum_pt, sum_p, sum_t) per block to ws.
__global__ __launch_bounds__(THREADS)
void cm_partial_kernel(const float* __restrict__ p,
                       const float* __restrict__ t,
                       float* __restrict__ ws) {
    const int plane = blockIdx.x >> 5;        // / BLOCKS_PER_PLANE
    const int sub   = blockIdx.x & 31;        // % BLOCKS_PER_PLANE
    const long base = (long)plane * PLANE_N + (long)sub * CHUNK;

    const v4f* __restrict__ p4 = (const v4f*)(p + base);
    const v4f* __restrict__ t4 = (const v4f*)(t + base);

    const int tid = threadIdx.x;

    float spt = 0.0f, sp = 0.0f, st = 0.0f;
#pragma unroll
    for (int i = 0; i < ITERS; ++i) {
        // Read-once data: non-temporal loads keep L2 clean.
        v4f a = __builtin_nontemporal_load(&p4[i * THREADS + tid]);
        v4f b = __builtin_nontemporal_load(&t4[i * THREADS + tid]);
        spt = fmaf(a[0], b[0], spt);
        spt = fmaf(a[1], b[1], spt);
        spt = fmaf(a[2], b[2], spt);
        spt = fmaf(a[3], b[3], spt);
        sp += (a[0] + a[1]) + (a[2] + a[3]);
        st += (b[0] + b[1]) + (b[2] + b[3]);
    }

    // Wave-level reduction on the CDNA5 matrix pipe.
    spt = wave_sum32_wmma(spt);
    sp  = wave_sum32_wmma(sp);
    st  = wave_sum32_wmma(st);

    __shared__ float red[THREADS / 32][3];
    const int wave = tid >> 5;
    if ((tid & 31) == 0) {
        red[wave][0] = spt;
        red[wave][1] = sp;
        red[wave][2] = st;
    }
    __syncthreads();

    if (tid == 0) {
        float a = 0.0f, b = 0.0f, c = 0.0f;
#pragma unroll
        for (int w = 0; w < THREADS / 32; ++w) {
            a += red[w][0];
            b += red[w][1];
            c += red[w][2];
        }
        float* o = ws + (long)blockIdx.x * 3;
        o[0] = a;
        o[1] = b;
        o[2] = c;
    }
}

// Kernel 2: one wave per plane reduces the 32 block-partials and emits
// [tp, tn, fp, fn].
__global__ __launch_bounds__(32)
void cm_final_kernel(const float* __restrict__ ws, float* __restrict__ out) {
    const int plane = blockIdx.x;
    const int lane  = threadIdx.x;

    const float* w = ws + ((long)plane * BLOCKS_PER_PLANE + lane) * 3;
    float spt = wave_sum32_wmma(w[0]);
    float sp  = wave_sum32_wmma(w[1]);
    float st  = wave_sum32_wmma(w[2]);

    if (lane == 0) {
        const float N = (float)PLANE_N;
        const float tp = spt;
        const float tn = N - sp - st + spt;
        const float fp = sp - spt;
        const float fn = st - spt;
        float* o = out + (long)plane * 4;
        o[0] = tp;
        o[1] = tn;
        o[2] = fp;
        o[3] = fn;
    }
}

extern "C" void kernel_launch(void* const* d_in, const int* in_sizes, int n_in,
                              void* d_out, int out_size, void* d_ws, size_t ws_size,
                              hipStream_t stream) {
    const float* y_pred = (const float*)d_in[0];
    const float* y_true = (const float*)d_in[1];
    float* out = (float*)d_out;
    float* ws  = (float*)d_ws;

    const int planes = in_sizes[0] / PLANE_N;            // 16*3 = 48
    const int nblocks = planes * BLOCKS_PER_PLANE;       // 1536

    cm_partial_kernel<<<nblocks, THREADS, 0, stream>>>(y_pred, y_true, ws);
    cm_final_kernel<<<planes, 32, 0, stream>>>(ws, out);
}